// DGCNN_66992899883199
// MI455X (gfx1250) — compile-verified
//
#include <hip/hip_runtime.h>
#include <hip/hip_bf16.h>

#define IN_DIM 9
#define HID    32
#define SORT_K 30
#define NG     2048

typedef __attribute__((ext_vector_type(2))) float v2f;
typedef __attribute__((ext_vector_type(8))) float v8f;

// ---------------- utility kernels ----------------

__global__ void fill_f32_kernel(float* p, float v, long long n) {
    long long i = (long long)blockIdx.x * blockDim.x + threadIdx.x;
    if (i < n) p[i] = v;
}

// deg accumulation: deg[dst[e]] += 1
__global__ void edge_deg_kernel(const int* __restrict__ dst, float* __restrict__ deg, long long E) {
    long long e = (long long)blockIdx.x * blockDim.x + threadIdx.x;
    if (e < E) atomicAdd(&deg[dst[e]], 1.0f);
}

__global__ void rsqrt_kernel(float* __restrict__ d, int n) {
    int i = blockIdx.x * blockDim.x + threadIdx.x;
    if (i < n) d[i] = rsqrtf(d[i]);
}

__global__ void count_kernel(const int* __restrict__ batch, int* __restrict__ counts, int n) {
    int i = blockIdx.x * blockDim.x + threadIdx.x;
    if (i < n) atomicAdd(&counts[batch[i]], 1);
}

__global__ void scan_kernel(const int* __restrict__ counts, int* __restrict__ starts) {
    __shared__ int sc[NG];
    for (int i = threadIdx.x; i < NG; i += blockDim.x) sc[i] = counts[i];
    __syncthreads();
    if (threadIdx.x == 0) {
        int run = 0;
        for (int g = 0; g < NG; g++) { int c = sc[g]; sc[g] = run; run += c; }
    }
    __syncthreads();
    for (int i = threadIdx.x; i < NG; i += blockDim.x) starts[i] = sc[i];
}

// ---------------- GCN layers ----------------

// h = x @ W1  (N x 9) @ (9 x 32), one thread per output element
__global__ void mm9_kernel(const float* __restrict__ x, const float* __restrict__ W,
                           float* __restrict__ out, int N) {
    __shared__ float sW[IN_DIM * HID];
    for (int i = threadIdx.x; i < IN_DIM * HID; i += blockDim.x) sW[i] = W[i];
    __syncthreads();
    long long idx = (long long)blockIdx.x * blockDim.x + threadIdx.x;
    if (idx >= (long long)N * HID) return;
    int row = (int)(idx >> 5), c = (int)(idx & 31);
    float s = 0.0f;
    const float* xr = x + (size_t)row * IN_DIM;
    #pragma unroll
    for (int k = 0; k < IN_DIM; k++) s += xr[k] * sW[k * HID + c];
    out[idx] = s;
}

// h = A @ W  (N x 32) @ (32 x 32) using V_WMMA_F32_16X16X4_F32.
// One wave per 16-row tile; two 16x16 accumulators cover 32 output columns.
__global__ void mm32_wmma_kernel(const float* __restrict__ A, const float* __restrict__ W,
                                 float* __restrict__ out, int N) {
    int wavesPerBlock = blockDim.x >> 5;
    int wave = blockIdx.x * wavesPerBlock + (threadIdx.x >> 5);
    int lane = threadIdx.x & 31;
    int row0 = wave * 16;
    if (row0 >= N) return;              // wave-uniform: EXEC stays all-ones for WMMA

    int m    = lane & 15;               // M (for A) / N (for B) index within half-wave
    int kgrp = (lane >> 4) << 1;        // lanes 0-15 -> K{0,1}; lanes 16-31 -> K{2,3}
    const float* arow = A + (size_t)(row0 + m) * HID;

    v8f c0 = {}; v8f c1 = {};
    #pragma unroll
    for (int kk = 0; kk < HID; kk += 4) {
        v2f a; a.x = arow[kk + kgrp]; a.y = arow[kk + kgrp + 1];
        v2f b0, b1;
        b0.x = W[(kk + kgrp)     * HID + m];
        b0.y = W[(kk + kgrp + 1) * HID + m];
        b1.x = W[(kk + kgrp)     * HID + 16 + m];
        b1.y = W[(kk + kgrp + 1) * HID + 16 + m];
        c0 = __builtin_amdgcn_wmma_f32_16x16x4_f32(false, a, false, b0, (short)0, c0, false, false);
        c1 = __builtin_amdgcn_wmma_f32_16x16x4_f32(false, a, false, b1, (short)0, c1, false, false);
    }
    // C/D layout: VGPR v holds row M = v + (lane<16 ? 0 : 8), col = lane&15
    int rowOff = (lane >> 4) << 3;
    #pragma unroll
    for (int v = 0; v < 8; v++) {
        out[(size_t)(row0 + rowOff + v) * HID + m]      = c0[v];
        out[(size_t)(row0 + rowOff + v) * HID + 16 + m] = c1[v];
    }
}

// agg[i] = h[i] * dis[i]^2  (self term; also initializes agg buffer)
__global__ void self_term_kernel(const float* __restrict__ h, const float* __restrict__ dis,
                                 float* __restrict__ agg, int N) {
    long long idx = (long long)blockIdx.x * blockDim.x + threadIdx.x;
    if (idx >= (long long)N * HID) return;
    float d = dis[idx >> 5];
    agg[idx] = h[idx] * d * d;
}

// agg[dst] += h[src] * dis[src] * dis[dst]  -- 8 threads per edge, 4 feats each
__global__ void edge_agg_kernel(const int* __restrict__ src, const int* __restrict__ dst,
                                const float* __restrict__ h, const float* __restrict__ dis,
                                float* __restrict__ agg, long long E) {
    long long t = (long long)blockIdx.x * blockDim.x + threadIdx.x;
    if (t >= E * 8) return;
    long long e = t >> 3;
    int fq = (int)(t & 7) << 2;
    int s = src[e], d = dst[e];
    float w = dis[s] * dis[d];
    float4 hv = *(const float4*)(h + (size_t)s * HID + fq);
    float* ap = agg + (size_t)d * HID + fq;
    atomicAdd(ap + 0, hv.x * w);
    atomicAdd(ap + 1, hv.y * w);
    atomicAdd(ap + 2, hv.z * w);
    atomicAdd(ap + 3, hv.w * w);
}

__global__ void bias_relu_kernel(float* __restrict__ h, const float* __restrict__ b, int N) {
    long long idx = (long long)blockIdx.x * blockDim.x + threadIdx.x;
    if (idx >= (long long)N * HID) return;
    h[idx] = fmaxf(h[idx] + b[idx & 31], 0.0f);
}

// ---------------- sort pool (one wave32 per graph) ----------------

__global__ void sort_pool_kernel(const float* __restrict__ h, const int* __restrict__ starts,
                                 const int* __restrict__ counts, float* __restrict__ pooled) {
    int g = blockIdx.x;
    int lane = threadIdx.x;
    int start = starts[g];
    int cnt = counts[g];
    int kk = cnt < SORT_K ? cnt : SORT_K;
    float prevKey = __builtin_inff();
    int prevIdx = -1;
    for (int j = 0; j < kk; j++) {
        float bestKey = -__builtin_inff();
        int bestIdx = 0x7fffffff;
        for (int i = start + lane; i < start + cnt; i += 32) {
            float key = h[(size_t)i * HID + (HID - 1)];
            // eligible = strictly after (prevKey, prevIdx) in (key desc, idx asc) order
            bool elig = (key < prevKey) || (key == prevKey && i > prevIdx);
            if (elig && (key > bestKey || (key == bestKey && i < bestIdx))) {
                bestKey = key; bestIdx = i;
            }
        }
        #pragma unroll
        for (int off = 16; off > 0; off >>= 1) {
            float ok = __shfl_xor(bestKey, off, 32);
            int   oi = __shfl_xor(bestIdx, off, 32);
            if (ok > bestKey || (ok == bestKey && oi < bestIdx)) { bestKey = ok; bestIdx = oi; }
        }
        pooled[((size_t)g * SORT_K + j) * HID + lane] = h[(size_t)bestIdx * HID + lane];
        prevKey = bestKey; prevIdx = bestIdx;
    }
}

// ---------------- conv / linear head ----------------

#define C1_T 26   // SORT_K - 4
#define C2_T 22   // C1_T - 4

__global__ void conv1_kernel(const float* __restrict__ pooled, const float* __restrict__ cw,
                             const float* __restrict__ cb, float* __restrict__ out) {
    __shared__ float sp[SORT_K * HID];       // 960 f32, [pos][feat]
    __shared__ float sw[16 * HID * 5];       // 2560 f32
    int g = blockIdx.x;
    for (int i = threadIdx.x; i < SORT_K * HID; i += blockDim.x)
        sp[i] = pooled[(size_t)g * SORT_K * HID + i];
    for (int i = threadIdx.x; i < 16 * HID * 5; i += blockDim.x) sw[i] = cw[i];
    __syncthreads();
    for (int oidx = threadIdx.x; oidx < 16 * C1_T; oidx += blockDim.x) {
        int o = oidx / C1_T, t = oidx % C1_T;
        float s = cb[o];
        for (int ic = 0; ic < HID; ic++)
            #pragma unroll
            for (int kt = 0; kt < 5; kt++)
                s += sp[(t + kt) * HID + ic] * sw[(o * HID + ic) * 5 + kt];
        out[(size_t)g * 16 * C1_T + oidx] = fmaxf(s, 0.0f);
    }
}

__global__ void conv2_kernel(const float* __restrict__ c1, const float* __restrict__ cw,
                             const float* __restrict__ cb, float* __restrict__ out) {
    __shared__ float si[16 * C1_T];          // [ic][t]
    __shared__ float sw[32 * 16 * 5];        // 2560 f32
    int g = blockIdx.x;
    for (int i = threadIdx.x; i < 16 * C1_T; i += blockDim.x)
        si[i] = c1[(size_t)g * 16 * C1_T + i];
    for (int i = threadIdx.x; i < 32 * 16 * 5; i += blockDim.x) sw[i] = cw[i];
    __syncthreads();
    for (int oidx = threadIdx.x; oidx < 32 * C2_T; oidx += blockDim.x) {
        int o = oidx / C2_T, t = oidx % C2_T;
        float s = cb[o];
        for (int ic = 0; ic < 16; ic++)
            #pragma unroll
            for (int kt = 0; kt < 5; kt++)
                s += si[ic * C1_T + t + kt] * sw[(o * 16 + ic) * 5 + kt];
        out[(size_t)g * 32 * C2_T + oidx] = fmaxf(s, 0.0f);
    }
}

__global__ void lin1_kernel(const float* __restrict__ c2, const float* __restrict__ lw,
                            const float* __restrict__ lb, float* __restrict__ q) {
    __shared__ float sf[32 * C2_T];          // 704 f32 flat features
    int g = blockIdx.x;
    for (int i = threadIdx.x; i < 32 * C2_T; i += blockDim.x)
        sf[i] = c2[(size_t)g * 32 * C2_T + i];
    __syncthreads();
    int j = threadIdx.x;                     // blockDim == 128
    float s = lb[j];
    for (int i = 0; i < 32 * C2_T; i++) s += sf[i] * lw[i * 128 + j];
    q[(size_t)g * 128 + j] = fmaxf(s, 0.0f);
}

__global__ void lin2_kernel(const float* __restrict__ q, const float* __restrict__ lw,
                            const float* __restrict__ lb, float* __restrict__ out, int G) {
    int g = blockIdx.x * blockDim.x + threadIdx.x;
    if (g >= G) return;
    float s = lb[0];
    for (int i = 0; i < 128; i++) s += q[(size_t)g * 128 + i] * lw[i];
    out[g] = s;
}

// ---------------- launch ----------------

extern "C" void kernel_launch(void* const* d_in, const int* in_sizes, int n_in,
                              void* d_out, int out_size, void* d_ws, size_t ws_size,
                              hipStream_t stream) {
    const float* x    = (const float*)d_in[0];
    const int*   ei   = (const int*)d_in[1];
    const int*   batch= (const int*)d_in[2];
    const float* W1   = (const float*)d_in[4];
    const float* b1   = (const float*)d_in[5];
    const float* W2   = (const float*)d_in[6];
    const float* b2   = (const float*)d_in[7];
    const float* W3   = (const float*)d_in[8];
    const float* b3   = (const float*)d_in[9];
    const float* cw1  = (const float*)d_in[10];
    const float* cb1  = (const float*)d_in[11];
    const float* cw2  = (const float*)d_in[12];
    const float* cb2  = (const float*)d_in[13];
    const float* lw1  = (const float*)d_in[14];
    const float* lb1  = (const float*)d_in[15];
    const float* lw2  = (const float*)d_in[16];
    const float* lb2  = (const float*)d_in[17];
    float* out = (float*)d_out;

    const int N = in_sizes[0] / IN_DIM;
    const long long E = (long long)in_sizes[1] / 2;
    const int* src = ei;
    const int* dst = ei + E;

    // workspace layout (256B aligned chunks)
    char* ws = (char*)d_ws;
    size_t off = 0;
    auto take = [&](size_t bytes) { char* p = ws + off; off += (bytes + 255) & ~(size_t)255; return p; };
    float* dis    = (float*)take((size_t)N * 4);
    float* B1     = (float*)take((size_t)N * HID * 4);
    float* B2     = (float*)take((size_t)N * HID * 4);
    int*   counts = (int*)  take((size_t)NG * 4);
    int*   starts = (int*)  take((size_t)NG * 4);
    float* pooled = (float*)take((size_t)NG * SORT_K * HID * 4);
    float* c1buf  = (float*)take((size_t)NG * 16 * C1_T * 4);
    float* c2buf  = (float*)take((size_t)NG * 32 * C2_T * 4);
    float* qbuf   = (float*)take((size_t)NG * 128 * 4);
    (void)ws_size; (void)n_in; (void)out_size;

    const int TB = 256;
    long long NH = (long long)N * HID;
    int gNH = (int)((NH + TB - 1) / TB);
    int gN  = (N + TB - 1) / TB;
    int gE  = (int)((E + TB - 1) / TB);
    int gE8 = (int)((E * 8 + TB - 1) / TB);

    // degree -> dis = rsqrt(deg+1)
    fill_f32_kernel<<<gN, TB, 0, stream>>>(dis, 1.0f, N);
    edge_deg_kernel<<<gE, TB, 0, stream>>>(dst, dis, E);
    rsqrt_kernel<<<gN, TB, 0, stream>>>(dis, N);

    // per-graph counts/starts (batch is sorted)
    fill_f32_kernel<<<(NG + TB - 1) / TB, TB, 0, stream>>>((float*)counts, 0.0f, NG);
    count_kernel<<<gN, TB, 0, stream>>>(batch, counts, N);
    scan_kernel<<<1, 256, 0, stream>>>(counts, starts);
    fill_f32_kernel<<<((long long)NG * SORT_K * HID + TB - 1) / TB, TB, 0, stream>>>(
        pooled, 0.0f, (long long)NG * SORT_K * HID);

    int tiles = N / 16;                       // 250000 % 16 == 0
    int mmBlocks = (tiles + 7) / 8;           // 8 waves (256 thr) per block

    // ---- layer 1: h in B2 ----
    mm9_kernel<<<gNH, TB, 0, stream>>>(x, W1, B1, N);
    self_term_kernel<<<gNH, TB, 0, stream>>>(B1, dis, B2, N);
    edge_agg_kernel<<<gE8, TB, 0, stream>>>(src, dst, B1, dis, B2, E);
    bias_relu_kernel<<<gNH, TB, 0, stream>>>(B2, b1, N);

    // ---- layer 2 ----
    mm32_wmma_kernel<<<mmBlocks, TB, 0, stream>>>(B2, W2, B1, N);
    self_term_kernel<<<gNH, TB, 0, stream>>>(B1, dis, B2, N);
    edge_agg_kernel<<<gE8, TB, 0, stream>>>(src, dst, B1, dis, B2, E);
    bias_relu_kernel<<<gNH, TB, 0, stream>>>(B2, b2, N);

    // ---- layer 3 ----
    mm32_wmma_kernel<<<mmBlocks, TB, 0, stream>>>(B2, W3, B1, N);
    self_term_kernel<<<gNH, TB, 0, stream>>>(B1, dis, B2, N);
    edge_agg_kernel<<<gE8, TB, 0, stream>>>(src, dst, B1, dis, B2, E);
    bias_relu_kernel<<<gNH, TB, 0, stream>>>(B2, b3, N);

    // ---- sort pool + conv head ----
    sort_pool_kernel<<<NG, 32, 0, stream>>>(B2, starts, counts, pooled);
    conv1_kernel<<<NG, 128, 0, stream>>>(pooled, cw1, cb1, c1buf);
    conv2_kernel<<<NG, 128, 0, stream>>>(c1buf, cw2, cb2, c2buf);
    lin1_kernel<<<NG, 128, 0, stream>>>(c2buf, lw1, lb1, qbuf);
    lin2_kernel<<<(NG + TB - 1) / TB, TB, 0, stream>>>(qbuf, lw2, lb2, out, NG);
}